// VehCollLoss_51608327028963
// MI455X (gfx1250) — compile-verified
//
#include <hip/hip_runtime.h>
#include <math.h>

// ---------------------------------------------------------------------------
// VehCollLoss on MI455X (gfx1250, wave32).
//
// Pairwise squared distance via Gram trick:
//   ||pi - pj||^2 = [xi, yi, ni, 1] . [-2xj, -2yj, 1, nj]^T   (ni = xi^2+yi^2)
// => one V_WMMA_F32_16X16X4_F32 per 16x16 agent tile per circle pair.
// 25 WMMAs + elementwise v_min per tile; min over circle pairs stays in VGPRs.
// Kernel is bound by the 6.6 MB output store (~0.3 us at 23.3 TB/s HBM).
// ---------------------------------------------------------------------------

#define NUM_CIRC 5

typedef __attribute__((ext_vector_type(2))) float v2f;
typedef __attribute__((ext_vector_type(8))) float v8f;

// --- prep: world circle centers [T][C][NA][2], radii [NA], scene ids [NA] ---
__global__ void prep_kernel(const float* __restrict__ traj,     // [NA][T][4]
                            const float* __restrict__ veh_att,  // [NA][2]
                            const int*   __restrict__ ptr,      // [nptr]
                            float* __restrict__ wc,             // [T][C][NA][2]
                            float* __restrict__ rad,            // [NA]
                            int*   __restrict__ sid,            // [NA]
                            int NA, int T, int nptr) {
  int idx = blockIdx.x * blockDim.x + threadIdx.x;

  if (idx < NA) {
    rad[idx] = veh_att[idx * 2 + 1] * 0.5f;
    int s = 0;  // searchsorted(ptr[1:], idx, side='right')
    for (int k = 1; k < nptr; ++k) s += (ptr[k] <= idx) ? 1 : 0;
    sid[idx] = s;
  }

  int total = T * NUM_CIRC * NA;
  if (idx >= total) return;

  int a = idx % NA;
  int c = (idx / NA) % NUM_CIRC;
  int t = idx / (NA * NUM_CIRC);

  const float4 tr = *(const float4*)(traj + ((size_t)a * T + t) * 4);
  float len = veh_att[a * 2 + 0];
  float wid = veh_att[a * 2 + 1];
  float r = wid * 0.5f, hl = len * 0.5f;
  float cmin = -hl + r, cmax = hl - r;
  float cx = cmin + (cmax - cmin) * ((float)c * (1.0f / (NUM_CIRC - 1)));
  float inv = 1.0f / sqrtf(tr.z * tr.z + tr.w * tr.w);  // normalize heading
  float wx = tr.x + cx * tr.z * inv;
  float wy = tr.y + cx * tr.w * inv;
  float2* o = (float2*)(wc + (((size_t)t * NUM_CIRC + c) * NA + a) * 2);
  *o = make_float2(wx, wy);
}

// --- main: one wave per 16x16 agent tile, 25 f32 WMMAs + min, mask, store ---
__global__ void __launch_bounds__(256)
coll_tile_kernel(const float* __restrict__ wc,   // [T][C][NA][2]
                 const float* __restrict__ rad,  // [NA]
                 const int*   __restrict__ sid,  // [NA]
                 float* __restrict__ out,        // [T][NA][NA]
                 int NA, int T) {
  constexpr int C = NUM_CIRC;
  int t    = blockIdx.y;
  int ti   = blockIdx.x;           // row tile
  int wave = threadIdx.x >> 5;     // column tile (8 waves -> 8 tiles = NA/16)
  int lane = threadIdx.x & 31;
  int ntiles = NA >> 4;
  if (wave >= ntiles) return;      // wave-uniform
  int tj  = wave;
  int gi0 = ti << 4, gj0 = tj << 4;

  const float* wct = wc + (size_t)t * C * NA * 2;
  __builtin_prefetch(wct, 0, 0);   // gfx1250 global_prefetch_b8

  int  lm = lane & 15;
  bool hi = lane >= 16;

  // A fragment (16x4 f32, 2 VGPRs): lanes 0-15 hold K0/K1, lanes 16-31 K2/K3.
  //   row m: [x, y | n, 1]
  // B fragment (4x16 f32, 2 VGPRs): col n: [-2x, -2y | 1, n]
  v2f A[C], B[C];
#pragma unroll
  for (int c = 0; c < C; ++c) {
    float2 p = *(const float2*)(wct + ((size_t)c * NA + gi0 + lm) * 2);
    float np = p.x * p.x + p.y * p.y;
    A[c].x = hi ? np : p.x;
    A[c].y = hi ? 1.0f : p.y;
    float2 q = *(const float2*)(wct + ((size_t)c * NA + gj0 + lm) * 2);
    float nq = q.x * q.x + q.y * q.y;
    B[c].x = hi ? 1.0f : (-2.0f * q.x);
    B[c].y = hi ? nq : (-2.0f * q.y);
  }

  v8f zero = {};
  v8f mn;
#pragma unroll
  for (int r = 0; r < 8; ++r) mn[r] = 3.402823466e+38f;

#pragma unroll
  for (int ci = 0; ci < C; ++ci) {
#pragma unroll
    for (int cj = 0; cj < C; ++cj) {
      // D = A x B + 0 : 16x16 tile of squared distances for circle pair (ci,cj)
      v8f d = __builtin_amdgcn_wmma_f32_16x16x4_f32(
          /*neg_a=*/false, A[ci], /*neg_b=*/false, B[cj],
          /*c_mod=*/(short)0, zero, /*reuse_a=*/false, /*reuse_b=*/false);
#pragma unroll
      for (int r = 0; r < 8; ++r) mn[r] = fminf(mn[r], d[r]);
    }
  }

  // D layout: VGPR r, lanes 0-15 -> row r, lanes 16-31 -> row r+8; col = lane&15
  float radj = rad[gj0 + lm];
  int   sidj = sid[gj0 + lm];
  int   gj   = gj0 + lm;
#pragma unroll
  for (int r = 0; r < 8; ++r) {
    int gi = gi0 + r + (hi ? 8 : 0);
    float sq   = fmaxf(mn[r], 0.0f);
    float dist = sqrtf(sq);
    float pd   = rad[gi] + radj;
    bool valid = (sid[gi] == sidj) && (gi != gj);
    float pen  = (valid && dist <= pd) ? (1.0f - dist / pd) : 0.0f;
    out[((size_t)t * NA + gi) * NA + gj] = pen;
  }
}

// --- scalar: num_pairs = sum(g^2 - g) over scenes ---
__global__ void npairs_kernel(const int* __restrict__ ptr, int nptr,
                              float* __restrict__ out, size_t off) {
  if (threadIdx.x == 0 && blockIdx.x == 0) {
    long long s = 0;
    for (int k = 0; k + 1 < nptr; ++k) {
      long long g = ptr[k + 1] - ptr[k];
      s += g * g - g;
    }
    out[off] = (float)s;
  }
}

extern "C" void kernel_launch(void* const* d_in, const int* in_sizes, int n_in,
                              void* d_out, int out_size, void* d_ws, size_t ws_size,
                              hipStream_t stream) {
  const float* traj    = (const float*)d_in[0];  // [NA][T][4]
  const float* veh_att = (const float*)d_in[1];  // [NA][2]
  const int*   ptr     = (const int*)d_in[2];    // [nptr]

  int NA   = in_sizes[1] / 2;            // 128
  int T    = in_sizes[0] / (NA * 4);     // 100
  int nptr = in_sizes[2];                // 9

  // workspace: wc [T][C][NA][2] f32, rad [NA] f32, sid [NA] i32
  float* wc  = (float*)d_ws;
  size_t wcElems = (size_t)T * NUM_CIRC * NA * 2;
  float* rad = wc + wcElems;
  int*   sid = (int*)(rad + NA);

  float* out = (float*)d_out;

  int total = T * NUM_CIRC * NA;
  prep_kernel<<<(total + 255) / 256, 256, 0, stream>>>(
      traj, veh_att, ptr, wc, rad, sid, NA, T, nptr);

  dim3 grid(NA / 16, T);  // 8 x 100 blocks, 8 waves/block -> one wave per tile
  coll_tile_kernel<<<grid, 256, 0, stream>>>(wc, rad, sid, out, NA, T);

  npairs_kernel<<<1, 32, 0, stream>>>(ptr, nptr, out, (size_t)T * NA * NA);
}